// OSDT_12936441495560
// MI455X (gfx1250) — compile-verified
//
#include <hip/hip_runtime.h>
#include <math.h>

typedef __attribute__((ext_vector_type(2))) float v2f;
typedef __attribute__((ext_vector_type(4))) float f4;
typedef __attribute__((ext_vector_type(8))) float v8f;

#define BATCH     16384
#define IN_DIM    512
#define DEPTH     10
#define NLEAF     1024
#define WPAD      514          // padded LDS row stride (64-bank conflict free, 8B aligned)
#define WAVES     8
#define ROWS_PER_WAVE  16
#define ROWS_PER_BLOCK (WAVES * ROWS_PER_WAVE)   // 128

__global__ __launch_bounds__(256, 2) void osdt_fused_kernel(
    const float* __restrict__ x,       // [16384][512]
    const float* __restrict__ w,       // [10][512]
    const float* __restrict__ thr,     // [10]
    const float* __restrict__ ltemp,   // [10]
    float* __restrict__ out)           // [16384][1024]
{
    __shared__ float wlds[16 * WPAD];                 // W padded to 16 rows
    __shared__ float Llds[WAVES][16][16];             // left  sigmoid per (row, depth)
    __shared__ float Rlds[WAVES][16][16];             // right sigmoid per (row, depth)

    const int tid  = threadIdx.x;
    const int lane = tid & 31;
    const int wave = tid >> 5;
    const int block_row0 = blockIdx.x * ROWS_PER_BLOCK;

    // ---- stage split_weights into LDS (rows >= DEPTH zeroed) ----
    for (int i = tid; i < 16 * IN_DIM; i += 256) {
        int r = i >> 9;          // / 512
        int k = i & (IN_DIM - 1);
        wlds[r * WPAD + k] = (r < DEPTH) ? w[r * IN_DIM + k] : 0.0f;
    }
    __syncthreads();

    // ---- Phase 1: selected = x @ W^T via V_WMMA_F32_16X16X4_F32 ----
    // A (16x4 f32): lanes 0-15 hold M=lane, K=k0/k0+1; lanes 16-31 hold M=lane-16, K=k0+2/k0+3
    // B (4x16 f32): lanes 0-15 hold N=lane, K=k0/k0+1; lanes 16-31 hold N=lane-16, K=k0+2/k0+3
    const int mrow  = lane & 15;
    const int khalf = (lane >> 4) << 1;   // 0 or 2
    const float* xrow = x + (size_t)(block_row0 + wave * ROWS_PER_WAVE + mrow) * IN_DIM;
    const float* wcol = &wlds[mrow * WPAD];   // B column N = lane&15 == row of W

    v8f acc = {};
    for (int k0 = 0; k0 < IN_DIM; k0 += 4) {
        v2f a, b;
        a.x = xrow[k0 + khalf];
        a.y = xrow[k0 + khalf + 1];
        b.x = wcol[k0 + khalf];
        b.y = wcol[k0 + khalf + 1];
        acc = __builtin_amdgcn_wmma_f32_16x16x4_f32(
            /*neg_a=*/false, a, /*neg_b=*/false, b,
            /*c_mod=*/(short)0, acc, /*reuse_a=*/false, /*reuse_b=*/false);
    }

    // ---- logits -> numerically-stable sigmoid pair, park in LDS ----
    // C layout: lane N = lane&15 -> depth d ; VGPR r -> M = r + 8*(lane>>4) -> batch row
    {
        const int d   = lane & 15;
        const float th = (d < DEPTH) ? thr[d]            : 0.0f;
        const float it = (d < DEPTH) ? __expf(-ltemp[d]) : 0.0f;
        const int mbase = (lane >> 4) << 3;
        #pragma unroll
        for (int r = 0; r < 8; ++r) {
            float z  = (acc[r] - th) * it;
            float t  = __expf(-fabsf(z));
            float bg = 1.0f / (1.0f + t);   // sigmoid(|z|)
            float sm = t * bg;              // sigmoid(-|z|)
            Rlds[wave][mbase + r][d] = (z >= 0.0f) ? bg : sm;  // sigmoid(z)
            Llds[wave][mbase + r][d] = (z >= 0.0f) ? sm : bg;  // sigmoid(-z)
        }
    }
    __syncthreads();

    // ---- Phase 2: out[b,l] = prod_d (bit_d(l) ? R[d] : L[d]) ----
    // leaf l = j*128 + lane*4 + i : bits0-1 = i, bits2-6 = lane, bits7-9 = j
    float* outbase = out + (size_t)(block_row0 + wave * ROWS_PER_WAVE) * NLEAF;
    for (int m = 0; m < 16; ++m) {
        const float* L = Llds[wave][m];   // broadcast LDS reads (same addr all lanes)
        const float* R = Rlds[wave][m];

        // depth 0,1 -> 4 pair products (the float4 store group)
        const float v00 = L[0] * L[1];
        const float v10 = R[0] * L[1];
        const float v01 = L[0] * R[1];
        const float v11 = R[0] * R[1];

        // depth 2..6 from lane bits (once per row)
        float base = (lane & 1)  ? R[2] : L[2];
        base *=      (lane & 2)  ? R[3] : L[3];
        base *=      (lane & 4)  ? R[4] : L[4];
        base *=      (lane & 8)  ? R[5] : L[5];
        base *=      (lane & 16) ? R[6] : L[6];

        float* orow = outbase + (size_t)m * NLEAF;
        #pragma unroll
        for (int j = 0; j < 8; ++j) {
            float mid = base;
            mid *= (j & 1) ? R[7] : L[7];
            mid *= (j & 2) ? R[8] : L[8];
            mid *= (j & 4) ? R[9] : L[9];
            f4 o = { v00 * mid, v10 * mid, v01 * mid, v11 * mid };
            *(f4*)(orow + (j << 7) + (lane << 2)) = o;   // coalesced b128 store
        }
    }
}

extern "C" void kernel_launch(void* const* d_in, const int* in_sizes, int n_in,
                              void* d_out, int out_size, void* d_ws, size_t ws_size,
                              hipStream_t stream) {
    const float* x     = (const float*)d_in[0];   // [16384*512]
    const float* w     = (const float*)d_in[1];   // [10*512]
    const float* thr   = (const float*)d_in[2];   // [10]
    const float* ltemp = (const float*)d_in[3];   // [10]
    float* out = (float*)d_out;                   // [16384*1024]

    dim3 grid(BATCH / ROWS_PER_BLOCK);            // 128 blocks
    dim3 block(256);                              // 8 waves
    osdt_fused_kernel<<<grid, block, 0, stream>>>(x, w, thr, ltemp, out);
}